// MAMBA_18657337933949
// MI455X (gfx1250) — compile-verified
//
#include <hip/hip_runtime.h>
#include <hip/hip_bf16.h>
#include <math.h>

#define D_MODEL 768
#define D_INNER 1536
#define D_STATE 16
#define D_CONV 4
#define DT_RANK 48
#define N_LAYER 3
#define BATCH 2
#define SEQLEN 1024
#define MROWS (BATCH * SEQLEN)   // 2048
#define EPS 1e-5f
#define XDBL_W (DT_RANK + 2 * D_STATE)   // 80

typedef __attribute__((ext_vector_type(2))) float v2f;
typedef __attribute__((ext_vector_type(4))) float v4f;
typedef __attribute__((ext_vector_type(8))) float v8f;
typedef __attribute__((ext_vector_type(4))) int   v4i;

// ---------------------------------------------------------------------------
// CDNA5 async global->LDS staging (ASYNCcnt path), with sync fallback.
// ---------------------------------------------------------------------------
#if __has_builtin(__builtin_amdgcn_global_load_async_to_lds_b128) && \
    __has_builtin(__builtin_amdgcn_s_wait_asynccnt)
#define USE_ASYNC_COPY 1
typedef __attribute__((address_space(1))) v4i gv4i;
typedef __attribute__((address_space(3))) v4i lv4i;
#endif

__device__ __forceinline__ void stage_b128(const float* g, float* l) {
#ifdef USE_ASYNC_COPY
    __builtin_amdgcn_global_load_async_to_lds_b128(
        (gv4i*)g, (lv4i*)l, 0, 0);
#else
    *(v4f*)l = *(const v4f*)g;
#endif
}

__device__ __forceinline__ void stage_wait() {
#ifdef USE_ASYNC_COPY
    __builtin_amdgcn_s_wait_asynccnt(0);
#endif
}

// ---------------------------------------------------------------------------
// Residual-add + RMSNorm.  One block per row (768 elems, 256 threads x 3).
// ---------------------------------------------------------------------------
__global__ __launch_bounds__(256) void mamba_resnorm(
    const float* __restrict__ hidden, const float* __restrict__ res_in,
    const float* __restrict__ w, float* __restrict__ res_out,
    float* __restrict__ normed)
{
    const int m = blockIdx.x;
    const int tid = threadIdx.x;
    __shared__ float red[256];

    float local[3];
    float ss = 0.f;
#pragma unroll
    for (int j = 0; j < 3; ++j) {
        int idx = tid + j * 256;
        float v = hidden[(size_t)m * D_MODEL + idx];
        if (res_in) v += res_in[(size_t)m * D_MODEL + idx];
        local[j] = v;
        ss += v * v;
    }
    red[tid] = ss;
    __syncthreads();
    for (int s = 128; s > 0; s >>= 1) {
        if (tid < s) red[tid] += red[tid + s];
        __syncthreads();
    }
    const float scale = rsqrtf(red[0] * (1.f / (float)D_MODEL) + EPS);
#pragma unroll
    for (int j = 0; j < 3; ++j) {
        int idx = tid + j * 256;
        res_out[(size_t)m * D_MODEL + idx] = local[j];
        normed[(size_t)m * D_MODEL + idx] = local[j] * scale * w[idx];
    }
}

// ---------------------------------------------------------------------------
// WMMA fp32 GEMM:  D[m][n] = sum_k A[m][k] * W[n][k]   (+ optional epilogue)
//   A: M x K row stride lda; W: N x K row-major; D: M x ldd.
// Block = 256 threads = 8 waves. Block tile 64(M) x 128(N), K slab 16.
// Wave grid 2(M) x 4(N); each wave computes a 32x32 tile as 2x2 WMMA
// accumulators (fragment reuse halves LDS traffic per WMMA).
// Tiles staged with GLOBAL_LOAD_ASYNC_TO_LDS_B128 when available.
// epilogue: 0 = none, 1 = softplus(x + bias[n]) (dt_proj path)
// ---------------------------------------------------------------------------
#define KT 16
#define LDSP 20   // padded LDS row stride in floats (80B: 16B aligned, no bank conflicts)

__global__ __launch_bounds__(256) void mamba_gemm_wmma(
    const float* __restrict__ A, int lda,
    const float* __restrict__ W,
    const float* __restrict__ bias,
    float* __restrict__ D, int ldd,
    int M, int N, int K, int epilogue)
{
    __shared__ float As[64 * LDSP];
    __shared__ float Bs[128 * LDSP];

    const int tid  = threadIdx.x;
    const int lane = tid & 31;
    const int wave = tid >> 5;
    const int wm   = wave >> 2;   // 0..1 : 32-row band
    const int wn   = wave & 3;    // 0..3 : 32-col band
    const int m0   = blockIdx.y * 64;
    const int n0   = blockIdx.x * 128;

    v8f acc[2][2];
#pragma unroll
    for (int mi = 0; mi < 2; ++mi)
#pragma unroll
        for (int ni = 0; ni < 2; ++ni)
            acc[mi][ni] = (v8f){0.f,0.f,0.f,0.f,0.f,0.f,0.f,0.f};

    // A tile 64x16 = 1024 floats -> one b128 per thread
    const int arow = tid >> 2;          // 0..63
    const int akk  = (tid & 3) * 4;     // 0,4,8,12 (16B aligned)

    const int r     = lane & 15;
    const int half  = (lane >> 4) << 1;   // K pair offset 0 or 2
    const int rbase = (lane >> 4) * 8;    // C/D row offset per lane half

    for (int k0 = 0; k0 < K; k0 += KT) {
        stage_b128(A + (size_t)(m0 + arow) * lda + k0 + akk,
                   &As[arow * LDSP + akk]);
#pragma unroll
        for (int t = 0; t < 2; ++t) {
            const int brow = arow + t * 64;     // 0..127
            if (n0 + brow < N) {
                stage_b128(W + (size_t)(n0 + brow) * K + k0 + akk,
                           &Bs[brow * LDSP + akk]);
            } else {
                *(v4f*)&Bs[brow * LDSP + akk] = (v4f){0.f,0.f,0.f,0.f};
            }
        }
        stage_wait();
        __syncthreads();

#pragma unroll
        for (int kk = 0; kk < KT; kk += 4) {
            v2f af[2], bf[2];
#pragma unroll
            for (int mi = 0; mi < 2; ++mi) {
                // A 16x4 frag: lanes 0-15 K={kk,kk+1}, lanes 16-31 K={kk+2,kk+3}
                af[mi].x = As[(wm * 32 + mi * 16 + r) * LDSP + kk + half];
                af[mi].y = As[(wm * 32 + mi * 16 + r) * LDSP + kk + half + 1];
            }
#pragma unroll
            for (int ni = 0; ni < 2; ++ni) {
                // B 4x16 frag mirrors A across lane halves; Bs rows are W rows
                bf[ni].x = Bs[(wn * 32 + ni * 16 + r) * LDSP + kk + half];
                bf[ni].y = Bs[(wn * 32 + ni * 16 + r) * LDSP + kk + half + 1];
            }
#pragma unroll
            for (int mi = 0; mi < 2; ++mi)
#pragma unroll
                for (int ni = 0; ni < 2; ++ni)
                    acc[mi][ni] = __builtin_amdgcn_wmma_f32_16x16x4_f32(
                        false, af[mi], false, bf[ni], (short)0, acc[mi][ni],
                        false, false);
        }
        __syncthreads();
    }

#pragma unroll
    for (int ni = 0; ni < 2; ++ni) {
        const int col = n0 + wn * 32 + ni * 16 + r;
        if (col < N) {
#pragma unroll
            for (int mi = 0; mi < 2; ++mi) {
#pragma unroll
                for (int v = 0; v < 8; ++v) {
                    int row = m0 + wm * 32 + mi * 16 + rbase + v;
                    float x = acc[mi][ni][v];
                    if (epilogue == 1) {
                        x += bias[col];
                        x = (x > 20.f) ? x : logf(1.f + __expf(x)); // softplus
                    }
                    D[(size_t)row * ldd + col] = x;
                }
            }
        }
    }
}

// ---------------------------------------------------------------------------
// Depthwise causal conv1d (width 4) + bias + SiLU over the x half of xz.
// ---------------------------------------------------------------------------
__global__ __launch_bounds__(256) void mamba_conv_silu(
    const float* __restrict__ xz, const float* __restrict__ cw,
    const float* __restrict__ cb, float* __restrict__ out)
{
    int idx = blockIdx.x * 256 + threadIdx.x;
    if (idx >= MROWS * D_INNER) return;
    int m = idx / D_INNER;
    int d = idx - m * D_INNER;
    int b = m / SEQLEN;
    int l = m - b * SEQLEN;

    float acc = cb[d];
#pragma unroll
    for (int j = 0; j < D_CONV; ++j) {
        int ll = l + j - (D_CONV - 1);
        if (ll >= 0)
            acc += cw[d * D_CONV + j] *
                   xz[(size_t)(b * SEQLEN + ll) * (2 * D_INNER) + d];
    }
    out[(size_t)m * D_INNER + d] = acc / (1.f + __expf(-acc));   // silu
}

// ---------------------------------------------------------------------------
// Selective scan, one state per lane: thread (b, d, n) runs one independent
// scalar recurrence; y = sum_n h_n C_n reduced across the 16-lane d-group
// with shfl_xor.  Writes y * silu(z) in place over u.  No LDS, no barriers.
// ---------------------------------------------------------------------------
__global__ __launch_bounds__(256) void mamba_scan(
    const float* __restrict__ delta, float* __restrict__ u,
    const float* __restrict__ xdbl, const float* __restrict__ xz,
    const float* __restrict__ A_log, const float* __restrict__ Dp)
{
    const int b = blockIdx.y;
    const int tid = threadIdx.x;
    const int n = tid & 15;
    const int d = blockIdx.x * 16 + (tid >> 4);

    const float a  = -__expf(A_log[(size_t)d * D_STATE + n]);
    const float Dv = Dp[d];
    float h = 0.f;

    for (int l = 0; l < SEQLEN; ++l) {
        const size_t m = (size_t)b * SEQLEN + l;
        const float dt = delta[m * D_INNER + d];
        const float ut = u[m * D_INNER + d];
        const float Bn = xdbl[m * XDBL_W + DT_RANK + n];
        const float Cn = xdbl[m * XDBL_W + DT_RANK + D_STATE + n];

        h = __expf(dt * a) * h + (dt * ut) * Bn;
        float p = h * Cn;
        p += __shfl_xor(p, 1, 16);
        p += __shfl_xor(p, 2, 16);
        p += __shfl_xor(p, 4, 16);
        p += __shfl_xor(p, 8, 16);
        if (n == 0) {
            float z = xz[m * (2 * D_INNER) + D_INNER + d];
            float y = p + ut * Dv;
            u[m * D_INNER + d] = y * (z / (1.f + __expf(-z)));
        }
    }
}

// ---------------------------------------------------------------------------
extern "C" void kernel_launch(void* const* d_in, const int* in_sizes, int n_in,
                              void* d_out, int out_size, void* d_ws, size_t ws_size,
                              hipStream_t stream) {
    const float* hs      = (const float*)d_in[0];
    const float* norm_w  = (const float*)d_in[1];
    const float* inp_w   = (const float*)d_in[2];
    const float* conv_w  = (const float*)d_in[3];
    const float* conv_b  = (const float*)d_in[4];
    const float* xprj_w  = (const float*)d_in[5];
    const float* dt_w    = (const float*)d_in[6];
    const float* dt_b    = (const float*)d_in[7];
    const float* A_log   = (const float*)d_in[8];
    const float* D_skip  = (const float*)d_in[9];
    const float* out_w   = (const float*)d_in[10];
    const float* normf_w = (const float*)d_in[11];
    float* out = (float*)d_out;

    // Workspace carve-up (floats)
    float* ws = (float*)d_ws;
    float* residual = ws;                                     // 2048*768
    float* normed   = residual + (size_t)MROWS * D_MODEL;     // 2048*768
    float* hiddenb  = normed   + (size_t)MROWS * D_MODEL;     // 2048*768
    float* xz       = hiddenb  + (size_t)MROWS * D_MODEL;     // 2048*3072
    float* xconv    = xz       + (size_t)MROWS * 2 * D_INNER; // 2048*1536
    float* xdbl     = xconv    + (size_t)MROWS * D_INNER;     // 2048*80
    float* delta    = xdbl     + (size_t)MROWS * XDBL_W;      // 2048*1536

    const dim3 blk(256);
    const dim3 g_gemm_in((2 * D_INNER + 127) / 128, MROWS / 64);  // N=3072
    const dim3 g_gemm_xp((XDBL_W + 127) / 128, MROWS / 64);       // N=80
    const dim3 g_gemm_dt((D_INNER + 127) / 128, MROWS / 64);      // N=1536
    const dim3 g_gemm_out((D_MODEL + 127) / 128, MROWS / 64);     // N=768
    const dim3 g_conv((MROWS * D_INNER + 255) / 256);
    const dim3 g_scan(D_INNER / 16, BATCH);                       // 96 x 2

    const float* hid = hs;
    const float* res = nullptr;

    for (int i = 0; i < N_LAYER; ++i) {
        mamba_resnorm<<<MROWS, blk, 0, stream>>>(
            hid, res, norm_w + (size_t)i * D_MODEL, residual, normed);
        res = residual;

        // in_proj: [2048x768] x [3072x768]^T -> xz
        mamba_gemm_wmma<<<g_gemm_in, blk, 0, stream>>>(
            normed, D_MODEL, inp_w + (size_t)i * 2 * D_INNER * D_MODEL, nullptr,
            xz, 2 * D_INNER, MROWS, 2 * D_INNER, D_MODEL, 0);

        mamba_conv_silu<<<g_conv, blk, 0, stream>>>(
            xz, conv_w + (size_t)i * D_INNER * D_CONV,
            conv_b + (size_t)i * D_INNER, xconv);

        // x_proj: [2048x1536] x [80x1536]^T -> x_dbl
        mamba_gemm_wmma<<<g_gemm_xp, blk, 0, stream>>>(
            xconv, D_INNER, xprj_w + (size_t)i * XDBL_W * D_INNER,
            nullptr, xdbl, XDBL_W, MROWS, XDBL_W, D_INNER, 0);

        // dt_proj + softplus: [2048x48] x [1536x48]^T -> delta
        mamba_gemm_wmma<<<g_gemm_dt, blk, 0, stream>>>(
            xdbl, XDBL_W, dt_w + (size_t)i * D_INNER * DT_RANK,
            dt_b + (size_t)i * D_INNER, delta, D_INNER,
            MROWS, D_INNER, DT_RANK, 1);

        // selective scan (in place: xconv becomes y * silu(z))
        mamba_scan<<<g_scan, blk, 0, stream>>>(
            delta, xconv, xdbl, xz,
            A_log + (size_t)i * D_INNER * D_STATE,
            D_skip + (size_t)i * D_INNER);

        // out_proj: [2048x1536] x [768x1536]^T -> hidden
        mamba_gemm_wmma<<<g_gemm_out, blk, 0, stream>>>(
            xconv, D_INNER, out_w + (size_t)i * D_MODEL * D_INNER, nullptr,
            hiddenb, D_MODEL, MROWS, D_MODEL, D_INNER, 0);
        hid = hiddenb;
    }

    // Final: rmsnorm(hidden + residual, norm_f_w) -> d_out
    mamba_resnorm<<<MROWS, blk, 0, stream>>>(
        hid, res, normf_w, residual, out);
}